// Sampler_44040594653458
// MI455X (gfx1250) — compile-verified
//
#include <hip/hip_runtime.h>
#include <hip/hip_bf16.h>
#include <stdint.h>

// ---------------- problem constants (match reference) ----------------
#define V_SIZE     128000
#define S_ROWS     256
#define PROMPT_LEN 1024
#define OUT_LEN    128
#define TOP_K      50
#define TOP_P      0.9f
#define REP_PEN    1.1f
#define INV_REP    (1.0f / 1.1f)
#define FREQ_PEN   0.1f
#define PRES_PEN   0.2f
#define INV_TEMP   1.25f          // 1 / 0.8

// ---------------- kernel tuning ----------------
#define NT     256                 // threads per block (8 wave32)
#define NB     4096                // histogram buckets = top 12 bits of key
#define CAP    2048                // candidate capacity (pow2 for bitonic)
#define TILE   1024                // floats per streamed tile = NT*4
#define NTILES (V_SIZE / TILE)     // 125 exactly
#define DEPTH  8                   // async pipeline depth (8 MB in flight device-wide)

// pen[] encoding: bit0 = prompt-mask, bit1 = out-mask,
// bits[23:8] = out count, bit30 = eos token
#define EOS_BIT (1 << 30)

// ---------------- CDNA5 async global->LDS helpers ----------------
__device__ __forceinline__ uint32_t lds_offset(const void* p) {
  // generic LDS pointer: low 32 bits are the wave-relative LDS byte offset
  return (uint32_t)(uintptr_t)p;
}

__device__ __forceinline__ void async_load_b128(uint32_t ldsoff, const void* gaddr) {
  // VGLOBAL async copy: memory -> LDS, tracked with ASYNCcnt (no VGPR data)
  asm volatile("global_load_async_to_lds_b128 %0, %1, off"
               :: "v"(ldsoff), "v"(gaddr) : "memory");
}

#define WAIT_ASYNC_LE(n) asm volatile("s_wait_asynccnt " #n ::: "memory")

// ---------------- math helpers ----------------
__device__ __forceinline__ float adjust_logit(float L, int p) {
  if (p & EOS_BIT) return -INFINITY;
  float x = L;
  if (p & 3) x = (L > 0.0f) ? (L * INV_REP) : (L * REP_PEN);
  x -= FREQ_PEN * (float)((p >> 8) & 0xFFFF);
  if (p & 2) x -= PRES_PEN;
  return x * INV_TEMP;
}

__device__ __forceinline__ uint32_t mono_key(float x) {
  uint32_t u = __float_as_uint(x);
  return (u & 0x80000000u) ? ~u : (u | 0x80000000u);
}

__device__ __forceinline__ float row_uniform(uint32_t row) {
  uint32_t h = row * 0x9E3779B9u + 0x7F4A7C15u;
  h ^= h >> 16; h *= 0x7FEB352Du;
  h ^= h >> 15; h *= 0x846CA68Bu;
  h ^= h >> 16;
  return (float)(h >> 8) * (1.0f / 16777216.0f);
}

// ---------------- kernel: per-row top-k/top-p sampler ----------------
__global__ __launch_bounds__(NT) void sampler_row_kernel(
    const float* __restrict__ logits, const int* __restrict__ pen,
    int* __restrict__ tok_out, float* __restrict__ probs_out) {
  __shared__ float stage[DEPTH][TILE];  // async-copy staging ring (32 KB)
  __shared__ int   hist[NB];
  __shared__ int   csum[NT];
  __shared__ float cval[CAP];
  __shared__ int   cidx[CAP];
  __shared__ int   sh_threshB, sh_count, sh_Ms;
  __shared__ float sh_m0, sh_inv;

  const int row = blockIdx.x;
  const int tid = threadIdx.x;
  const float* __restrict__ rowp = logits + (size_t)row * V_SIZE;
  float* __restrict__ prow = probs_out + (size_t)row * V_SIZE;

  // ---- kick off the async pipeline FIRST: tiles 0..6 in flight while we
  // clear the histogram (stage ring and hist are disjoint LDS regions) ----
  const uint32_t my_base = lds_offset(&stage[0][tid * 4]);
#pragma unroll
  for (int t = 0; t < DEPTH - 1; ++t)
    async_load_b128(my_base + (uint32_t)t * (TILE * 4), rowp + t * TILE + tid * 4);

  for (int i = tid; i < NB; i += NT) hist[i] = 0;
  if (tid == 0) { sh_threshB = NB - 1; sh_count = 0; }
  __syncthreads();

  // ---- phase 1: 8-deep async-pipelined stream + 12-bit key histogram ----
  // Each thread consumes exactly the 16 B it issued, so per-wave ASYNCcnt
  // waits are sufficient (no barrier inside the pipeline).
  for (int t = 0; t < NTILES; ++t) {
    const int buf = t & (DEPTH - 1);
    if (t + DEPTH - 1 < NTILES) {
      async_load_b128(my_base + (uint32_t)((t + DEPTH - 1) & (DEPTH - 1)) * (TILE * 4),
                      rowp + (t + DEPTH - 1) * TILE + tid * 4);
      WAIT_ASYNC_LE(7);               // in-order: tile t has landed
    } else {
      WAIT_ASYNC_LE(0);               // drain at the tail
    }
    const int v0 = t * TILE + tid * 4;
    const int4   p4 = ((const int4*)pen)[v0 >> 2];
    const float4 x4 = *(const float4*)&stage[buf][tid * 4];
    const float xs_[4] = {x4.x, x4.y, x4.z, x4.w};
    const int   ps_[4] = {p4.x, p4.y, p4.z, p4.w};
#pragma unroll
    for (int c = 0; c < 4; ++c) {
      float x = adjust_logit(xs_[c], ps_[c]);
      atomicAdd(&hist[mono_key(x) >> 20], 1);
    }
  }
  __syncthreads();

  // ---- find bucket containing the TOP_K-th largest element ----
  {
    int s = 0;
#pragma unroll
    for (int b = 0; b < NB / NT; ++b) s += hist[tid * (NB / NT) + b];
    csum[tid] = s;
    __syncthreads();
    if (tid == 0) {                      // exclusive suffix sums of 256 chunks
      int acc = 0;
      for (int t2 = NT - 1; t2 >= 0; --t2) { int c = csum[t2]; csum[t2] = acc; acc += c; }
    }
    __syncthreads();
    int cum = csum[tid];                 // elements strictly above my chunk
    for (int b = NB / NT - 1; b >= 0; --b) {
      int idx = tid * (NB / NT) + b;
      int c = hist[idx];
      if (cum < TOP_K && cum + c >= TOP_K) sh_threshB = idx;   // unique crossing
      cum += c;
    }
    __syncthreads();
  }
  const uint32_t tb = (uint32_t)sh_threshB;

  // ---- zero this row of probs (must complete before survivor scatter) ----
  float4 z4 = make_float4(0.f, 0.f, 0.f, 0.f);
  for (int i = tid; i < V_SIZE / 4; i += NT) ((float4*)prow)[i] = z4;

  // ---- phase 2: collect candidates (131 MB logits are L2-resident: 192 MB L2) ----
  for (int i = tid * 4; i < V_SIZE; i += NT * 4) {
    __builtin_prefetch(rowp + i + NT * 16, 0, 0);   // global_prefetch
    const float4 x4 = *(const float4*)(rowp + i);
    const int4   p4 = *(const int4*)(pen + i);
    const float xs_[4] = {x4.x, x4.y, x4.z, x4.w};
    const int   ps_[4] = {p4.x, p4.y, p4.z, p4.w};
#pragma unroll
    for (int c = 0; c < 4; ++c) {
      float x = adjust_logit(xs_[c], ps_[c]);
      if ((mono_key(x) >> 20) >= tb) {
        int pos = atomicAdd(&sh_count, 1);
        if (pos < CAP) { cval[pos] = x; cidx[pos] = i + c; }
      }
    }
  }
  __syncthreads();
  const int n = (sh_count < CAP) ? sh_count : CAP;
  for (int i = tid; i < CAP; i += NT)
    if (i >= n) { cval[i] = -INFINITY; cidx[i] = -1; }
  __syncthreads();

  // ---- bitonic sort candidates descending (CAP = 2048) ----
  for (int k = 2; k <= CAP; k <<= 1) {
    for (int j = k >> 1; j > 0; j >>= 1) {
      for (int i = tid; i < CAP; i += NT) {
        int ixj = i ^ j;
        if (ixj > i) {
          float a = cval[i], b = cval[ixj];
          bool up = ((i & k) == 0);
          if (up ? (a < b) : (a > b)) {
            cval[i] = b; cval[ixj] = a;
            int ti = cidx[i]; cidx[i] = cidx[ixj]; cidx[ixj] = ti;
          }
        }
      }
      __syncthreads();
    }
  }

  // ---- exact top-k threshold, top-p cut, renormalize, sample ----
  if (tid == 0) {
    int eff = (n < TOP_K) ? n : TOP_K;
    float kth = cval[eff - 1];
    int M = eff;
    while (M < n && cval[M] >= kth) ++M;       // keep ties at the k-th value
    float m0 = cval[0];
    float total = 0.f;
    for (int j = 0; j < M; ++j) total += __expf(cval[j] - m0);
    float lim = TOP_P * total;
    float pre = 0.f; int Ms = 0;               // survivor j iff prefix_excl < lim
    while (Ms < M && pre < lim) { pre += __expf(cval[Ms] - m0); ++Ms; }
    float ssum = pre;
    float u = row_uniform((uint32_t)row) * ssum;
    int pick = Ms - 1; float cacc = 0.f;
    for (int j = 0; j < Ms; ++j) {
      cacc += __expf(cval[j] - m0);
      if (u <= cacc) { pick = j; break; }
    }
    tok_out[row] = cidx[pick];
    sh_Ms = Ms; sh_m0 = m0; sh_inv = 1.0f / ssum;
  }
  __syncthreads();

  // ---- scatter the <=~50 nonzero renormalized probabilities ----
  for (int j = tid; j < sh_Ms; j += NT)
    prow[cidx[j]] = __expf(cval[j] - sh_m0) * sh_inv;
}

// ---------------- kernel: zero penalty table ----------------
__global__ __launch_bounds__(NT) void zero_pen_kernel(int4* __restrict__ pen4) {
  int i = blockIdx.x * NT + threadIdx.x;
  if (i < V_SIZE / 4) pen4[i] = make_int4(0, 0, 0, 0);
}

// ---------------- kernel: scatter token penalties ----------------
__global__ __launch_bounds__(NT) void scatter_pen_kernel(
    int* __restrict__ pen, const int* __restrict__ prompt,
    const int* __restrict__ outtok, const int* __restrict__ eos) {
  for (int t = threadIdx.x; t < PROMPT_LEN; t += NT)
    atomicOr(&pen[prompt[t]], 1);
  for (int t = threadIdx.x; t < OUT_LEN; t += NT) {
    int v = outtok[t];
    atomicOr(&pen[v], 2);
    atomicAdd(&pen[v], 256);
  }
  for (int t = threadIdx.x; t < 2; t += NT)
    atomicOr(&pen[eos[t]], EOS_BIT);
}

// ---------------- host entry ----------------
extern "C" void kernel_launch(void* const* d_in, const int* in_sizes, int n_in,
                              void* d_out, int out_size, void* d_ws, size_t ws_size,
                              hipStream_t stream) {
  (void)in_sizes; (void)n_in; (void)out_size; (void)ws_size;
  const float* logits = (const float*)d_in[0];   // [1,S,V] f32
  const int*   prompt = (const int*)d_in[1];     // [1,1024] i32
  const int*   outtok = (const int*)d_in[2];     // [128]    i32
  const int*   eos    = (const int*)d_in[3];     // [2]      i32

  int* pen = (int*)d_ws;                         // V * 4 bytes = 512 KB scratch

  // output: (sampled_tokens[1,S] i32, probs[S,V] f32) concatenated flat
  int*   tok_out = (int*)d_out;                  // int32 bits in first S slots
  float* probs   = (float*)d_out + S_ROWS;

  zero_pen_kernel<<<(V_SIZE / 4 + NT - 1) / NT, NT, 0, stream>>>((int4*)pen);
  scatter_pen_kernel<<<1, NT, 0, stream>>>(pen, prompt, outtok, eos);
  sampler_row_kernel<<<S_ROWS, NT, 0, stream>>>(logits, pen, tok_out, probs);
}